// LossVariance_49374944034948
// MI455X (gfx1250) — compile-verified
//
#include <hip/hip_runtime.h>
#include <stdint.h>

// Problem constants (match reference)
#define NUM_LABELS 500
#define LBINS      512                    // padded bin count (covers 500)
#define NBATCH     8
#define NPIX       (1024*1024)            // pixels per image
#define TPB        256                    // 8 wave32 waves per block
#define ITERS      4                      // float4 iterations per thread
#define VECS_PER_BLOCK (TPB*ITERS)        // 1024 x float4/int4
#define PIX_PER_BLOCK  (VECS_PER_BLOCK*4) // 4096 pixels
#define BLOCKS_PER_BATCH (NPIX/PIX_PER_BLOCK) // 256

typedef unsigned int v4u __attribute__((ext_vector_type(4)));
typedef int          v4i __attribute__((ext_vector_type(4)));
typedef int          v8i __attribute__((ext_vector_type(8)));

#if defined(__has_builtin)
#  if __has_builtin(__builtin_amdgcn_tensor_load_to_lds) && \
      __has_builtin(__builtin_amdgcn_s_wait_tensorcnt)
#    define HAVE_TDM 1
#  endif
#endif
#ifndef HAVE_TDM
#  define HAVE_TDM 0
#endif

// ---------------------------------------------------------------------------
// Kernel 0: zero the global accumulator (d_ws is poisoned with 0xAA).
// Layout per batch b: [0..511]=sum, [512..1023]=sumsq, [1024..1535]=cnt
// ---------------------------------------------------------------------------
__global__ void lv_zero(float* __restrict__ acc) {
    int i = blockIdx.x * blockDim.x + threadIdx.x;
    if (i < NBATCH * 3 * LBINS) acc[i] = 0.0f;
}

// ---------------------------------------------------------------------------
// Kernel 1: binned accumulation.
//  - wave 0 DMAs the block's 16KB label tile into LDS via TENSOR_LOAD_TO_LDS
//    (async, TENSORcnt) while all waves zero the LDS histogram
//  - x streamed with coalesced global_load_b128 + global_prefetch_b8
//  - per-pixel ds_add_f32 binning; global_atomic_add_f32 flush of live bins
// ---------------------------------------------------------------------------
__device__ __forceinline__ void bin_add(float* __restrict__ sh_s,
                                        float* __restrict__ sh_ss,
                                        float* __restrict__ sh_c,
                                        int l, float a, float b, float c) {
    float v  = a + b + c;
    float vv = a*a + b*b + c*c;
    atomicAdd(&sh_s[l],  v);    // ds_add_f32 (no return used)
    atomicAdd(&sh_ss[l], vv);   // ds_add_f32
    atomicAdd(&sh_c[l],  1.0f); // ds_add_f32
}

__global__ __launch_bounds__(TPB) void lv_accum(const float* __restrict__ x,
                                                const int*   __restrict__ tgt,
                                                float*       __restrict__ acc) {
    __shared__ float sh_s [LBINS];
    __shared__ float sh_ss[LBINS];
    __shared__ float sh_c [LBINS];
    __shared__ int   sh_tgt[PIX_PER_BLOCK];   // 16 KB label tile

    const int t = threadIdx.x;
    const int b = blockIdx.y;
    const size_t tile_pix = (size_t)blockIdx.x * PIX_PER_BLOCK;
    const int* __restrict__ tg_glob = tgt + (size_t)b * NPIX + tile_pix;

#if HAVE_TDM
    if (t < 32) {   // wave 0 issues the tensor DMA (EXEC ignored by TDM)
        const uint64_t gaddr = (uint64_t)(uintptr_t)tg_glob;      // byte address
        const uint32_t laddr = (uint32_t)(uintptr_t)&sh_tgt[0];   // LDS byte offset

        // D# group 0: count=1 | lds_addr | global_addr[56:0] | type=2
        v4u g0;
        g0[0] = 1u;                                               // count=1, user mode
        g0[1] = laddr;                                            // lds_addr
        g0[2] = (uint32_t)(gaddr & 0xffffffffu);                  // global_addr[31:0]
        g0[3] = (uint32_t)((gaddr >> 32) & 0x01ffffffu)           // global_addr[56:32]
              | 0x80000000u;                                      // type=2 ("image")

        // D# group 1: data_size=4B, 2-D tensor 4096x1, tile 4096x1, stride 4096
        v8i g1;
        g1[0] = (int)(2u << 16);                                  // data_size=2 -> 4B
        g1[1] = (int)((uint32_t)PIX_PER_BLOCK << 16);             // tensor_dim0[15:0]
        g1[2] = (int)(1u << 16);                                  // tensor_dim1=1
        g1[3] = (int)((uint32_t)PIX_PER_BLOCK << 16);             // tile_dim0=4096
        g1[4] = 1;                                                // tile_dim1=1
        g1[5] = PIX_PER_BLOCK;                                    // tensor_dim0_stride
        g1[6] = 0;
        g1[7] = 0;

        v4i gz = {0, 0, 0, 0};                                    // groups 2/3 (2-D)
#if __clang_major__ >= 23
        v8i gz8 = {0, 0, 0, 0, 0, 0, 0, 0};
        __builtin_amdgcn_tensor_load_to_lds(g0, g1, gz, gz, gz8, 0);
#else
        __builtin_amdgcn_tensor_load_to_lds(g0, g1, gz, gz, 0);
#endif
    }
#endif

    // Zero the LDS histogram while the DMA is in flight
    #pragma unroll
    for (int i = t; i < LBINS; i += TPB) { sh_s[i] = 0.f; sh_ss[i] = 0.f; sh_c[i] = 0.f; }

#if HAVE_TDM
    if (t < 32) __builtin_amdgcn_s_wait_tensorcnt(0);   // s_wait_tensorcnt 0
#else
    {   // fallback: cooperative copy of the label tile
        const int4* __restrict__ tg4 = (const int4*)tg_glob;
        for (int i = t; i < VECS_PER_BLOCK; i += TPB) ((int4*)sh_tgt)[i] = tg4[i];
    }
#endif
    __syncthreads();

    const int planeVec = NPIX / 4;
    const float4* __restrict__ xb = (const float4*)(x + (size_t)b * 3 * NPIX);
    const float4* __restrict__ x0 = xb + (tile_pix / 4);
    const float4* __restrict__ x1 = x0 + planeVec;
    const float4* __restrict__ x2 = x1 + planeVec;
    const int4*   __restrict__ lt = (const int4*)sh_tgt;

    #pragma unroll
    for (int i = 0; i < ITERS; ++i) {
        const int vi = i * TPB + t;
        if (i + 1 < ITERS) {
            // gfx1250 prefetch path: global_prefetch_b8 for next iteration
            __builtin_prefetch(&x0[vi + TPB], 0, 1);
            __builtin_prefetch(&x1[vi + TPB], 0, 1);
            __builtin_prefetch(&x2[vi + TPB], 0, 1);
        }
        const float4 a  = x0[vi];      // global_load_b128, coalesced
        const float4 c1 = x1[vi];
        const float4 c2 = x2[vi];
        const int4   l  = lt[vi];      // ds_load_b128 from DMA'd tile

        bin_add(sh_s, sh_ss, sh_c, l.x, a.x, c1.x, c2.x);
        bin_add(sh_s, sh_ss, sh_c, l.y, a.y, c1.y, c2.y);
        bin_add(sh_s, sh_ss, sh_c, l.z, a.z, c1.z, c2.z);
        bin_add(sh_s, sh_ss, sh_c, l.w, a.w, c1.w, c2.w);
    }
    __syncthreads();

    float* __restrict__ out = acc + (size_t)b * 3 * LBINS;
    #pragma unroll
    for (int i = t; i < LBINS; i += TPB) {
        const float c = sh_c[i];
        if (c != 0.0f) {
            // agent-scope relaxed fp32 adds -> native global_atomic_add_f32
            __hip_atomic_fetch_add(&out[i],             sh_s[i],  __ATOMIC_RELAXED, __HIP_MEMORY_SCOPE_AGENT);
            __hip_atomic_fetch_add(&out[LBINS + i],     sh_ss[i], __ATOMIC_RELAXED, __HIP_MEMORY_SCOPE_AGENT);
            __hip_atomic_fetch_add(&out[2 * LBINS + i], c,        __ATOMIC_RELAXED, __HIP_MEMORY_SCOPE_AGENT);
        }
    }
}

// ---------------------------------------------------------------------------
// Kernel 2: finalize. One block of 512 threads (16 waves): per batch compute
// var per label (labels 1..499 only), reduce, divide by n_unique, mean.
// ---------------------------------------------------------------------------
__global__ __launch_bounds__(512) void lv_finalize(const float* __restrict__ acc,
                                                   float* __restrict__ out) {
    __shared__ float vsum;
    __shared__ float ucnt;
    __shared__ float total;
    const int t = threadIdx.x;
    if (t == 0) total = 0.0f;

    for (int b = 0; b < NBATCH; ++b) {
        if (t == 0) { vsum = 0.0f; ucnt = 0.0f; }
        __syncthreads();

        const float* __restrict__ a = acc + (size_t)b * 3 * LBINS;
        float var = 0.0f, u = 0.0f;
        if (t >= 1 && t < NUM_LABELS) {            // drop label 0, like reference
            const float s  = a[t];
            const float ss = a[LBINS + t];
            const float c  = a[2 * LBINS + t];
            if (c > 0.0f) u = 1.0f;
            if (c > 1.0f) {                        // valid = cnt > 1
                const float N = 3.0f * c;          // C * cnt, C == 3
                var = (ss - s * s / N) / (N - 1.0f);
            }
        }
        atomicAdd(&vsum, var);                     // ds_add_f32
        atomicAdd(&ucnt, u);
        __syncthreads();
        if (t == 0) total += vsum / (ucnt + 1e-8f);
        __syncthreads();
    }
    if (t == 0) out[0] = total / (float)NBATCH;
}

// ---------------------------------------------------------------------------
extern "C" void kernel_launch(void* const* d_in, const int* in_sizes, int n_in,
                              void* d_out, int out_size, void* d_ws, size_t ws_size,
                              hipStream_t stream) {
    const float* x   = (const float*)d_in[0];   // (8,3,1024,1024) f32
    const int*   tgt = (const int*)  d_in[1];   // (8,1024,1024) i32
    float* acc = (float*)d_ws;                  // 8*3*512 floats = 48 KB
    float* out = (float*)d_out;                 // 1 float

    const int accN = NBATCH * 3 * LBINS;
    lv_zero<<<(accN + 255) / 256, 256, 0, stream>>>(acc);

    dim3 grid(BLOCKS_PER_BATCH, NBATCH);
    lv_accum<<<grid, TPB, 0, stream>>>(x, tgt, acc);

    lv_finalize<<<1, 512, 0, stream>>>(acc, out);
}